// RasterizePointsXYsBlending_24498493456682
// MI455X (gfx1250) — compile-verified
//
#include <hip/hip_runtime.h>

// ---- problem constants (from the reference) ----
#define BB   4
#define NN   100000
#define CC   64
#define SS   256
#define KK   8
#define R_NDC    (1.5f / 256.0f * 2.0f)          // 0.01171875
#define R2       (R_NDC * R_NDC)
#define INV_R2   (1.0f / R2)
#define SENT     0xFFFFFFFFFFFFFFFFull

typedef float v2f __attribute__((ext_vector_type(2)));
typedef float v8f __attribute__((ext_vector_type(8)));

// ---------------------------------------------------------------------------
// Kernel 1: re-initialize per-pixel top-K slot buffer to sentinel every call
// slots: BB*SS*SS*KK u64 = 16 MB (lives in d_ws, L2-resident on MI455X)
// ---------------------------------------------------------------------------
__global__ __launch_bounds__(512) void rp_init_slots(unsigned long long* slots) {
    int gid = blockIdx.x * 512 + threadIdx.x;     // 4096 blocks * 512 = 2,097,152
    slots[gid] = SENT;
}

// ---------------------------------------------------------------------------
// Kernel 2: scatter points -> per-pixel top-8 via cascaded atomicMin(u64).
// key = (zbits << 32) | point_idx : ascending key order == (z asc, idx asc),
// exactly the reference's lexsort tie-break. The min/carry cascade conserves
// the key multiset, so final slots are exactly the 8 smallest keys per pixel.
// ---------------------------------------------------------------------------
__global__ __launch_bounds__(256) void rp_scatter(const float* __restrict__ pts,
                                                  unsigned long long* __restrict__ slots) {
    int gid = blockIdx.x * 256 + threadIdx.x;
    if (gid >= BB * NN) return;
    int b = gid / NN;
    int i = gid - b * NN;

    const float* p = pts + (size_t)gid * 3;
    float x = -p[0];                 // pts * [-1,-1,1]
    float y = -p[1];
    float z =  p[2];
    if (!(z >= 0.0f)) return;

    float px = (1.0f - x) * (SS * 0.5f) - 0.5f;
    float py = (1.0f - y) * (SS * 0.5f) - 0.5f;
    int jx = (int)floorf(px);
    int iy = (int)floorf(py);
    if (jx + 2 < 0 || jx - 1 > SS - 1 || iy + 2 < 0 || iy - 1 > SS - 1) return;

    unsigned long long key = ((unsigned long long)__float_as_uint(z) << 32) |
                             (unsigned long long)(unsigned)i;

    #pragma unroll
    for (int di = -1; di <= 2; ++di) {
        int ci = iy + di;
        if (ci < 0 || ci > SS - 1) continue;
        float cy = 1.0f - (2.0f * ci + 1.0f) / (float)SS;
        float dy = cy - y;
        float dy2 = dy * dy;
        #pragma unroll
        for (int dj = -1; dj <= 2; ++dj) {
            int cj = jx + dj;
            if (cj < 0 || cj > SS - 1) continue;
            float cx = 1.0f - (2.0f * cj + 1.0f) / (float)SS;
            float dx = cx - x;
            float d2 = dy2 + dx * dx;
            if (d2 > R2) continue;

            unsigned long long* slot =
                slots + (((size_t)b * SS * SS + (size_t)ci * SS + cj) << 3);
            unsigned long long cur = key;
            #pragma unroll
            for (int s = 0; s < KK; ++s) {
                unsigned long long prev = atomicMin(&slot[s], cur);
                unsigned long long mx = prev > cur ? prev : cur;   // displaced value
                if (mx == SENT) break;                             // absorbed into empty slot
                cur = mx;
            }
        }
    }
}

// ---------------------------------------------------------------------------
// Kernel 3: blend. One wave per 4x4 pixel tile.
//   Phase 1 (lanes 0-15): load 8 keys, sort asc, recompute d2 -> alpha ->
//                         transmittance weights; stash weights + feature-row
//                         BYTE offsets (pid*256) in LDS.
//   Phase 2 (all lanes):  out_tile(16x64) = W(16x128, block-diag) x F(128x64)
//                         as 32 K-steps x 4 N-chunks of V_WMMA_F32_16X16X4_F32.
//                         Per lane, A is nonzero only at t==2m and t==2m+1 with
//                         fixed slot positions -> 4 preloaded registers + selects
//                         (no LDS traffic for A in the loop). B gathers use a
//                         uniform base + 32-bit VGPR offset (GVS addressing),
//                         coalesced 64B across lanes 0-15 / 16-31.
// A layout (ISA 16x4 f32): lane m(+16), VGPR v -> A[m][2*(lane>=16)+v].
// B layout assumed mirrored: lane n(+16), VGPR v -> B[2*(lane>=16)+v][n].
// EXEC is all-ones at every WMMA (exact grid, uniform wave control flow).
// ---------------------------------------------------------------------------
#define WAVES 8
__global__ __launch_bounds__(256) void rp_blend(const float* __restrict__ pts,
                                                const float* __restrict__ feat,
                                                const unsigned long long* __restrict__ slots,
                                                float* __restrict__ out) {
    __shared__ float w_lds[WAVES][16][KK];
    __shared__ __align__(16) int o_lds[WAVES][128];   // byte offsets pid*CC*4

    int lane = threadIdx.x & 31;
    int wv   = threadIdx.x >> 5;
    int tile = blockIdx.x * WAVES + wv;          // 2048 blocks * 8 = 16384 tiles exactly
    int b    = tile >> 12;                       // 4096 tiles per batch
    int tl   = tile & 4095;
    int h0   = (tl >> 6) << 2;
    int w0   = (tl & 63) << 2;

    // -------- phase 1: per-pixel prep (lanes 0..15) --------
    if (lane < 16) {
        int m = lane;
        int h = h0 + (m >> 2);
        int w = w0 + (m & 3);
        const unsigned long long* sl =
            slots + (((size_t)b * SS * SS + (size_t)h * SS + w) << 3);
        unsigned long long a[KK];
        #pragma unroll
        for (int j = 0; j < KK; ++j) a[j] = sl[j];
        // sort ascending (fully unrolled -> stays in registers)
        #pragma unroll
        for (int i = 0; i < KK; ++i) {
            #pragma unroll
            for (int j = 0; j < KK - 1; ++j) {
                if (a[j] > a[j + 1]) {
                    unsigned long long t = a[j]; a[j] = a[j + 1]; a[j + 1] = t;
                }
            }
        }
        float cx = 1.0f - (2.0f * w + 1.0f) / (float)SS;
        float cy = 1.0f - (2.0f * h + 1.0f) / (float)SS;
        float T = 1.0f;
        #pragma unroll
        for (int j = 0; j < KK; ++j) {
            int   pidv = 0;
            float wj   = 0.0f;
            if (a[j] != SENT) {
                pidv = (int)(a[j] & 0xFFFFFFFFull);
                const float* pp = pts + ((size_t)b * NN + (size_t)pidv) * 3;
                float x = -pp[0];
                float y = -pp[1];
                float dx = cx - x;
                float dy = cy - y;
                float d2 = dx * dx + dy * dy;
                float dist = fminf(fmaxf(d2 * INV_R2, 0.001f), 1.0f);
                float alpha = 1.0f - sqrtf(dist);      // GAMMA = 1 -> pow(.,0.5)
                wj = alpha * T;
                T *= (1.0f - alpha);
            }
            w_lds[wv][m][j] = wj;
            o_lds[wv][m * KK + j] = pidv * (CC * 4);   // feature-row byte offset
        }
    }
    __syncthreads();

    // -------- phase 2: WMMA GEMM over 128 slots --------
    int m    = lane & 15;
    int half = lane >> 4;
    const char* featb = (const char*)(feat + (size_t)b * NN * CC);  // uniform base

    // Per-lane A values: nonzero only at t==2m (slots j=2*half,2*half+1)
    // and t==2m+1 (slots j=4+2*half,5+2*half).
    float wa0 = w_lds[wv][m][2 * half];
    float wa1 = w_lds[wv][m][2 * half + 1];
    float wb0 = w_lds[wv][m][4 + 2 * half];
    float wb1 = w_lds[wv][m][5 + 2 * half];
    int tA = 2 * m;
    int tB = 2 * m + 1;

    for (int chunk = 0; chunk < 4; ++chunk) {
        v8f acc = {0.f, 0.f, 0.f, 0.f, 0.f, 0.f, 0.f, 0.f};
        int ch4 = (chunk * 16 + (lane & 15)) * 4;      // channel byte offset
        #pragma unroll
        for (int t = 0; t < 32; ++t) {
            int s0 = 4 * t + 2 * half;                 // even -> 8B-aligned pair
            int2 off = *(const int2*)&o_lds[wv][s0];   // ds_load_b64
            v2f Bv;  // coalesced: lanes 0-15 read 16 consecutive channels of one row
            Bv.x = *(const float*)(featb + (unsigned)(off.x + ch4));
            Bv.y = *(const float*)(featb + (unsigned)(off.y + ch4));
            v2f A;
            A.x = (t == tA) ? wa0 : ((t == tB) ? wb0 : 0.0f);
            A.y = (t == tA) ? wa1 : ((t == tB) ? wb1 : 0.0f);
#if __has_builtin(__builtin_amdgcn_wmma_f32_16x16x4_f32)
            acc = __builtin_amdgcn_wmma_f32_16x16x4_f32(
                /*neg_a=*/false, A, /*neg_b=*/false, Bv,
                /*c_mod=*/(short)0, acc, /*reuse_a=*/false, /*reuse_b=*/false);
#else
            // scalar fallback with the same C/D lane layout
            #pragma unroll
            for (int r = 0; r < 8; ++r) {
                int mm = r + 8 * half;
                int mp0 = s0 >> 3, mp1 = (s0 + 1) >> 3;
                float s = 0.0f;
                if (mp0 == mm) s += w_lds[wv][mp0][s0 & 7] * Bv.x;
                if (mp1 == mm) s += w_lds[wv][mp1][(s0 + 1) & 7] * Bv.y;
                acc[r] += s;
            }
#endif
        }
        // store D: VGPR r, lanes 0-15 -> M=r, lanes 16-31 -> M=r+8; N = lane&15
        #pragma unroll
        for (int r = 0; r < 8; ++r) {
            int mm = r + 8 * half;
            int c  = chunk * 16 + (lane & 15);
            int h  = h0 + (mm >> 2);
            int w  = w0 + (mm & 3);
            out[(((size_t)b * CC + c) * SS + h) * SS + w] = acc[r];
        }
    }
}

// ---------------------------------------------------------------------------
extern "C" void kernel_launch(void* const* d_in, const int* in_sizes, int n_in,
                              void* d_out, int out_size, void* d_ws, size_t ws_size,
                              hipStream_t stream) {
    (void)in_sizes; (void)n_in; (void)out_size; (void)ws_size;
    const float* pts  = (const float*)d_in[0];   // (B,N,3) f32
    const float* feat = (const float*)d_in[1];   // (B,N,C) f32
    float*       out  = (float*)d_out;           // (B,C,S,S) f32
    unsigned long long* slots = (unsigned long long*)d_ws;   // 16 MB

    // BB*SS*SS*KK = 2,097,152 u64 sentinels
    rp_init_slots<<<(BB * SS * SS * KK) / 512, 512, 0, stream>>>(slots);
    rp_scatter<<<(BB * NN + 255) / 256, 256, 0, stream>>>(pts, slots);
    rp_blend<<<(BB * (SS / 4) * (SS / 4)) / WAVES, 256, 0, stream>>>(pts, feat, slots, out);
}